// GNNEncoder_58471684767804
// MI455X (gfx1250) — compile-verified
//
#include <hip/hip_runtime.h>
#include <hip/hip_bf16.h>

typedef __attribute__((ext_vector_type(2))) float v2f;
typedef __attribute__((ext_vector_type(8))) float v8f;

#define NNODES 50000
#define NEDGES 600000
#define DIM    128
#define TM     80    // rows of A per block (50000 / 80 = 625 blocks exactly)
#define LDA    132   // padded LDS stride in floats (multiple of 4, not mult of 64)

// ---------- degree / dinv ----------
__global__ void deg_init_kernel(float* __restrict__ deg, int n) {
  int i = blockIdx.x * blockDim.x + threadIdx.x;
  if (i < n) deg[i] = 1.0f;  // self-loop
}

__global__ void deg_count_kernel(const int* __restrict__ dst,
                                 float* __restrict__ deg, int e) {
  int i = blockIdx.x * blockDim.x + threadIdx.x;
  if (i < e) atomicAdd(&deg[dst[i]], 1.0f);
}

__global__ void rsqrt_kernel(float* __restrict__ deg, int n) {
  int i = blockIdx.x * blockDim.x + threadIdx.x;
  if (i < n) deg[i] = rsqrtf(deg[i]);
}

// ---------- XW = X @ W via fp32 WMMA (16x16x4) ----------
// Block: 256 threads = 8 wave32s. Block covers TM rows x 128 cols of output.
// Wave w owns column tile w (16 cols); loops over TM/16 row subtiles.
__global__ __launch_bounds__(256) void gemm_xw_wmma(
    const float* __restrict__ X, const float* __restrict__ W,
    float* __restrict__ XW) {
  __shared__ float As[TM * LDA];

  const int tid  = threadIdx.x;
  const int row0 = blockIdx.x * TM;

  // Cooperative coalesced load of A tile (TM x 128) into LDS as float4s.
  for (int c = tid; c < TM * (DIM / 4); c += 256) {
    int r  = c >> 5;       // c / 32
    int c4 = c & 31;       // float4 index within row
    float4 v = ((const float4*)(X + (size_t)(row0 + r) * DIM))[c4];
    *(float4*)(&As[r * LDA + c4 * 4]) = v;
  }
  __syncthreads();

  const int wave = tid >> 5;      // 0..7 -> column tile
  const int lane = tid & 31;
  const int half = lane >> 4;     // 0: K pair 0,1   1: K pair 2,3
  const int l    = lane & 15;
  const int ncol = wave * 16 + l;

  for (int mt = 0; mt < TM / 16; ++mt) {
    v8f acc = {};
    const float* arow = &As[(mt * 16 + l) * LDA];
    for (int k0 = 0; k0 < DIM; k0 += 4) {
      // A 16x4: lane(l,half) holds row l, K = k0+2*half (+1)
      v2f a, b;
      a.x = arow[k0 + 2 * half];
      a.y = arow[k0 + 2 * half + 1];
      // B 4x16: lane(l,half) holds col ncol, K = k0+2*half (+1)
      b.x = W[(size_t)(k0 + 2 * half) * DIM + ncol];
      b.y = W[(size_t)(k0 + 2 * half + 1) * DIM + ncol];
      acc = __builtin_amdgcn_wmma_f32_16x16x4_f32(
          /*neg_a=*/false, a, /*neg_b=*/false, b,
          /*c_mod=*/(short)0, acc, /*reuse_a=*/false, /*reuse_b=*/false);
    }
    // D 16x16 f32: VGPR v -> M = v + 8*half, N = ncol
    float* obase = XW + (size_t)(row0 + mt * 16 + 8 * half) * DIM + ncol;
#pragma unroll
    for (int v = 0; v < 8; ++v) {
      obase[(size_t)v * DIM] = acc[v];
    }
  }
}

// ---------- out = xw * dinv^2 + bias (self-loop + bias init) ----------
__global__ void accinit_kernel(const float* __restrict__ xw,
                               const float* __restrict__ dinv,
                               const float* __restrict__ bias,
                               float* __restrict__ out, int total) {
  int i = blockIdx.x * blockDim.x + threadIdx.x;
  if (i < total) {
    int node = i >> 7;     // / DIM
    int d    = i & 127;
    float di = dinv[node];
    out[i] = xw[i] * di * di + bias[d];
  }
}

// ---------- edge scatter: out[dst] += xw[src] * dinv[src]*dinv[dst] ----------
// One wave per edge; each lane moves 4 contiguous floats (32*4 = 128).
__global__ __launch_bounds__(256) void scatter_kernel(
    const int* __restrict__ src, const int* __restrict__ dst,
    const float* __restrict__ dinv, const float* __restrict__ xw,
    float* __restrict__ acc, int e) {
  int gid  = blockIdx.x * blockDim.x + threadIdx.x;
  int edge = gid >> 5;
  int lane = gid & 31;
  if (edge >= e) return;
  int s = src[edge];
  int d = dst[edge];
  float norm = dinv[s] * dinv[d];
  float4 v = ((const float4*)(xw + (size_t)s * DIM))[lane];
  float* o = acc + (size_t)d * DIM + lane * 4;
  atomicAdd(o + 0, v.x * norm);
  atomicAdd(o + 1, v.y * norm);
  atomicAdd(o + 2, v.z * norm);
  atomicAdd(o + 3, v.w * norm);
}

// ---------- in-place ReLU ----------
__global__ void relu_kernel(float* __restrict__ x, int total) {
  int i = blockIdx.x * blockDim.x + threadIdx.x;
  if (i < total) x[i] = fmaxf(x[i], 0.0f);
}

extern "C" void kernel_launch(void* const* d_in, const int* in_sizes, int n_in,
                              void* d_out, int out_size, void* d_ws, size_t ws_size,
                              hipStream_t stream) {
  const float* x   = (const float*)d_in[0];
  const int*   ei  = (const int*)d_in[1];       // [2, E]: row0 = src, row1 = dst
  const float* W1  = (const float*)d_in[2];
  const float* b1  = (const float*)d_in[3];
  const float* W2  = (const float*)d_in[4];
  const float* b2  = (const float*)d_in[5];
  float* out = (float*)d_out;

  const int* src = ei;
  const int* dst = ei + NEDGES;

  // workspace: [dinv: NNODES floats][xw: NNODES*DIM floats]
  float* dinv = (float*)d_ws;
  float* xw   = (float*)((char*)d_ws + (1 << 18));  // 256 KB offset, aligned

  const int total = NNODES * DIM;
  dim3 b256(256);

  // degree -> dinv
  deg_init_kernel<<<dim3((NNODES + 255) / 256), b256, 0, stream>>>(dinv, NNODES);
  deg_count_kernel<<<dim3((NEDGES + 255) / 256), b256, 0, stream>>>(dst, dinv, NEDGES);
  rsqrt_kernel<<<dim3((NNODES + 255) / 256), b256, 0, stream>>>(dinv, NNODES);

  // ---- layer 1 ----
  gemm_xw_wmma<<<dim3(NNODES / TM), b256, 0, stream>>>(x, W1, xw);
  accinit_kernel<<<dim3((total + 255) / 256), b256, 0, stream>>>(xw, dinv, b1, out, total);
  scatter_kernel<<<dim3((NEDGES * 32) / 256), b256, 0, stream>>>(src, dst, dinv, xw, out, NEDGES);
  relu_kernel<<<dim3((total + 255) / 256), b256, 0, stream>>>(out, total);

  // ---- layer 2 (out currently holds h; gemm reads it, then out is reused) ----
  gemm_xw_wmma<<<dim3(NNODES / TM), b256, 0, stream>>>(out, W2, xw);
  accinit_kernel<<<dim3((total + 255) / 256), b256, 0, stream>>>(xw, dinv, b2, out, total);
  scatter_kernel<<<dim3((NEDGES * 32) / 256), b256, 0, stream>>>(src, dst, dinv, xw, out, NEDGES);
  relu_kernel<<<dim3((total + 255) / 256), b256, 0, stream>>>(out, total);
}